// PointToMeshResidual_58145267253566
// MI455X (gfx1250) — compile-verified
//
#include <hip/hip_runtime.h>

#define BATCH 2
#define NFACE 8192
#define NQ    4096
#define WAVES_PER_BLOCK 8
#define TRIS_PER_WAVE (NFACE / WAVES_PER_BLOCK)   // 1024
#define TILES_PER_WAVE (TRIS_PER_WAVE / 16)       // 64

typedef __attribute__((ext_vector_type(2))) float v2f;
typedef __attribute__((ext_vector_type(8))) float v8f;

__device__ __forceinline__ float sdiv(float n, float d) {
  return n / (d == 0.0f ? 1.0f : d);
}

// Ericson region select, branchless (priority chain matches jnp.select order).
// Returns barycentric (v, w) along edges e1 = b-a, e2 = c-a; u = 1 - v - w.
__device__ __forceinline__ void bary_vw(float d1, float d2, float d3, float d4,
                                        float d5, float d6, float& v, float& w) {
  float vc = d1 * d4 - d3 * d2;
  float vb = d5 * d2 - d1 * d6;
  float va = d3 * d6 - d5 * d4;
  float v_ab = sdiv(d1, d1 - d3);
  float w_ac = sdiv(d2, d2 - d6);
  float w_bc = sdiv(d4 - d3, (d4 - d3) + (d5 - d6));
  float inv  = sdiv(1.0f, va + vb + vc);
  v = vb * inv;                     // interior default
  w = vc * inv;
  bool c6 = (va <= 0.f) && (d4 >= d3) && (d5 >= d6);
  v = c6 ? 1.0f - w_bc : v;  w = c6 ? w_bc : w;
  bool c5 = (vb <= 0.f) && (d2 >= 0.f) && (d6 <= 0.f);
  v = c5 ? 0.0f : v;         w = c5 ? w_ac : w;
  bool c4 = (d6 >= 0.f) && (d5 <= d6);
  v = c4 ? 0.0f : v;         w = c4 ? 1.0f : w;
  bool c3 = (vc <= 0.f) && (d1 >= 0.f) && (d3 <= 0.f);
  v = c3 ? v_ab : v;         w = c3 ? 0.0f : w;
  bool c2 = (d3 >= 0.f) && (d4 <= d3);
  v = c2 ? 1.0f : v;         w = c2 ? 0.0f : w;
  bool c1 = (d1 <= 0.f) && (d2 <= 0.f);
  v = c1 ? 0.0f : v;         w = c1 ? 0.0f : w;
}

// Stage 1: exact closest-triangle via WMMA-accelerated brute force.
// One wave = 16 points x (F/8) triangles; 8 waves per block cover all F.
// Triangle vertex fetch is software-pipelined one tile ahead.
__global__ __launch_bounds__(256) void stage1_closest(
    const float* __restrict__ tris, const float* __restrict__ pts,
    int* __restrict__ bestFace) {
  __shared__ float s_dist[WAVES_PER_BLOCK][16];
  __shared__ int   s_idx [WAVES_PER_BLOCK][16];

  const int lane = threadIdx.x & 31;
  const int wave = threadIdx.x >> 5;
  const int half = lane >> 4;   // which 16-lane half
  const int col  = lane & 15;   // column slot: triangle (B/D) and point (A) index
  const bool lo  = (half == 0);

  const int blocksPerBatch = NQ / 16;
  const int b     = blockIdx.x / blocksPerBatch;
  const int qBase = (blockIdx.x % blocksPerBatch) * 16;

  // A fragment: 16x4 f32, rows = points, K = (x, y, z, 0).
  // Layout: lanes 0-15 hold K=0 (v0) / K=1 (v1); lanes 16-31 hold K=2 / K=3.
  const float* P = pts + ((size_t)b * NQ + qBase) * 3;
  float px = P[col * 3 + 0];
  float py = P[col * 3 + 1];
  float pz = P[col * 3 + 2];

  v2f afrag;
  afrag.x = lo ? px : pz;
  afrag.y = lo ? py : 0.0f;

  // |p_m|^2 broadcast: lane m (m = col) computed it; each lane needs points
  // m = r + 8*half, which live in lanes 0..15 of this wave.
  float ppSelf = px * px + py * py + pz * pz;
  float pp[8];
#pragma unroll
  for (int r = 0; r < 8; ++r) pp[r] = __shfl(ppSelf, r + 8 * half, 32);

  float bd[8];
  int   bi[8];
#pragma unroll
  for (int r = 0; r < 8; ++r) { bd[r] = 3.0e38f; bi[r] = 0x7fffffff; }

  const float* TB = tris + (size_t)b * NFACE * 9;
  const int triBase = wave * TRIS_PER_WAVE;

  // --- software pipeline: preload tile 0 ---
  int gCur = triBase + col;
  {
    const float* tp = TB + (size_t)gCur * 9;
  }
  float t0, t1, t2, t3, t4, t5, t6, t7, t8;
  {
    const float* tp = TB + (size_t)gCur * 9;
    t0 = tp[0]; t1 = tp[1]; t2 = tp[2];
    t3 = tp[3]; t4 = tp[4]; t5 = tp[5];
    t6 = tp[6]; t7 = tp[7]; t8 = tp[8];
  }

  for (int tile = 0; tile < TILES_PER_WAVE; ++tile) {
    // prefetch next tile's 9 floats (clamped on the last iteration)
    const int gNext = (tile + 1 < TILES_PER_WAVE) ? (gCur + 16) : gCur;
    const float* tpn = TB + (size_t)gNext * 9;
    float n0 = tpn[0], n1 = tpn[1], n2 = tpn[2];
    float n3 = tpn[3], n4 = tpn[4], n5 = tpn[5];
    float n6 = tpn[6], n7 = tpn[7], n8 = tpn[8];

    // ---- compute on current tile (registers t0..t8, index gCur) ----
    float ax = t0, ay = t1, az = t2;
    float bx = t3, by = t4, bz = t5;
    float cx = t6, cy = t7, cz = t8;
    float e1x = bx - ax, e1y = by - ay, e1z = bz - az;
    float e2x = cx - ax, e2y = cy - ay, e2z = cz - az;
    float cc1 = e1x * ax + e1y * ay + e1z * az;   // e1 . a
    float cc2 = e2x * ax + e2y * ay + e2z * az;   // e2 . a
    float d11 = e1x * e1x + e1y * e1y + e1z * e1z;
    float d22 = e2x * e2x + e2y * e2y + e2z * e2z;
    float d12 = e1x * e2x + e1y * e2y + e1z * e2z;
    float aa  = ax * ax + ay * ay + az * az;

    // B fragments: 4x16 f32, columns = triangles, K rows = (x, y, z, 0).
    v2f bf1, bf2, bf3;
    bf1.x = lo ? e1x : e1z;  bf1.y = lo ? e1y : 0.0f;
    bf2.x = lo ? e2x : e2z;  bf2.y = lo ? e2y : 0.0f;
    bf3.x = lo ? ax  : az;   bf3.y = lo ? ay  : 0.0f;

    v8f zero = {};
    v8f S = __builtin_amdgcn_wmma_f32_16x16x4_f32(false, afrag, false, bf1,
                                                  (short)0, zero, false, false);
    v8f T = __builtin_amdgcn_wmma_f32_16x16x4_f32(false, afrag, false, bf2,
                                                  (short)0, zero, false, false);
    v8f U = __builtin_amdgcn_wmma_f32_16x16x4_f32(false, afrag, false, bf3,
                                                  (short)0, zero, false, false);

    // D layout: lane holds column gCur, rows (points) m = r + 8*half.
#pragma unroll
    for (int r = 0; r < 8; ++r) {
      float s = S[r], t = T[r], u = U[r];
      float d1 = s - cc1, d2 = t - cc2;
      float d3 = d1 - d11, d4 = d2 - d12;
      float d5 = d1 - d12, d6 = d2 - d22;
      float v, w;
      bary_vw(d1, d2, d3, d4, d5, d6, v, w);
      float app  = pp[r] - 2.0f * u + aa;  // |p - a|^2
      float dist = app - 2.0f * (v * d1 + w * d2) +
                   v * v * d11 + 2.0f * (v * w) * d12 + w * w * d22;
      if (dist < bd[r]) { bd[r] = dist; bi[r] = gCur; }
    }

    // ---- rotate pipeline registers ----
    t0 = n0; t1 = n1; t2 = n2; t3 = n3; t4 = n4;
    t5 = n5; t6 = n6; t7 = n7; t8 = n8;
    gCur = gNext;
  }

  // Reduce min+argmin across the 16 triangle columns (stays within each half).
#pragma unroll
  for (int off = 1; off < 16; off <<= 1) {
#pragma unroll
    for (int r = 0; r < 8; ++r) {
      float od = __shfl_xor(bd[r], off, 32);
      int   oi = __shfl_xor(bi[r], off, 32);
      if (od < bd[r] || (od == bd[r] && oi < bi[r])) { bd[r] = od; bi[r] = oi; }
    }
  }

  if (col == 0) {
#pragma unroll
    for (int r = 0; r < 8; ++r) {
      s_dist[wave][r + 8 * half] = bd[r];
      s_idx [wave][r + 8 * half] = bi[r];
    }
  }
  __syncthreads();

  // Cross-wave reduction (waves cover ascending triangle ranges).
  if (threadIdx.x < 16) {
    float best = s_dist[0][threadIdx.x];
    int   idx  = s_idx [0][threadIdx.x];
    for (int wv = 1; wv < WAVES_PER_BLOCK; ++wv) {
      float d = s_dist[wv][threadIdx.x];
      int   i = s_idx [wv][threadIdx.x];
      if (d < best || (d == best && i < idx)) { best = d; idx = i; }
    }
    bestFace[(size_t)b * NQ + qBase + threadIdx.x] = idx;
  }
}

// Stage 2: gather + barycentric interpolation epilogue (one thread per point).
__global__ __launch_bounds__(256) void stage2_gather(
    const float* __restrict__ tris, const float* __restrict__ pts,
    const float* __restrict__ nrms, const float* __restrict__ cmps,
    const int* __restrict__ faces, const int* __restrict__ bestFace,
    float* __restrict__ out) {
  int tid = blockIdx.x * blockDim.x + threadIdx.x;
  if (tid >= BATCH * NQ) return;
  int b = tid / NQ;
  int f = bestFace[tid];

  const float* tp = tris + ((size_t)b * NFACE + f) * 9;
  float ax = tp[0], ay = tp[1], az = tp[2];
  float bx = tp[3], by = tp[4], bz = tp[5];
  float cx = tp[6], cy = tp[7], cz = tp[8];
  const float* pq = pts + (size_t)tid * 3;
  float px = pq[0], py = pq[1], pz = pq[2];

  float e1x = bx - ax, e1y = by - ay, e1z = bz - az;
  float e2x = cx - ax, e2y = cy - ay, e2z = cz - az;
  float apx = px - ax, apy = py - ay, apz = pz - az;
  float bpx = px - bx, bpy = py - by, bpz = pz - bz;
  float cpx = px - cx, cpy = py - cy, cpz = pz - cz;
  float d1 = e1x * apx + e1y * apy + e1z * apz;
  float d2 = e2x * apx + e2y * apy + e2z * apz;
  float d3 = e1x * bpx + e1y * bpy + e1z * bpz;
  float d4 = e2x * bpx + e2y * bpy + e2z * bpz;
  float d5 = e1x * cpx + e1y * cpy + e1z * cpz;
  float d6 = e2x * cpx + e2y * cpy + e2z * cpz;
  float v, w;
  bary_vw(d1, d2, d3, d4, d5, d6, v, w);
  float u = 1.0f - v - w;
  u = fminf(fmaxf(u, 0.0f), 1.0f);
  v = fminf(fmaxf(v, 0.0f), 1.0f);
  w = fminf(fmaxf(w, 0.0f), 1.0f);

  float rx = u * ax + v * bx + w * cx - px;
  float ry = u * ay + v * by + w * cy - py;
  float rz = u * az + v * bz + w * cz - pz;

  const float* np_ = nrms + ((size_t)b * NFACE + f) * 9;
  float nx = u * np_[0] + v * np_[3] + w * np_[6];
  float ny = u * np_[1] + v * np_[4] + w * np_[7];
  float nz = u * np_[2] + v * np_[5] + w * np_[8];

  const float* mp = cmps + ((size_t)b * NFACE + f) * 9;
  float mx = u * mp[0] + v * mp[3] + w * mp[6];
  float my = u * mp[1] + v * mp[4] + w * mp[7];
  float mz = u * mp[2] + v * mp[5] + w * mp[8];

  // argmax-first over clipped (u, v, w)
  int k = (v > u) ? ((w > v) ? 2 : 1) : ((w > u) ? 2 : 0);
  int fid = faces[((size_t)b * NFACE + f) * 3 + k];

  const int N3 = BATCH * NQ * 3;
  out[0 * N3 + tid * 3 + 0] = rx;
  out[0 * N3 + tid * 3 + 1] = ry;
  out[0 * N3 + tid * 3 + 2] = rz;
  out[1 * N3 + tid * 3 + 0] = nx;
  out[1 * N3 + tid * 3 + 1] = ny;
  out[1 * N3 + tid * 3 + 2] = nz;
  out[2 * N3 + tid * 3 + 0] = mx;
  out[2 * N3 + tid * 3 + 1] = my;
  out[2 * N3 + tid * 3 + 2] = mz;
  out[3 * N3 + tid] = (float)fid;
}

extern "C" void kernel_launch(void* const* d_in, const int* in_sizes, int n_in,
                              void* d_out, int out_size, void* d_ws, size_t ws_size,
                              hipStream_t stream) {
  const float* tris  = (const float*)d_in[0];
  const float* pts   = (const float*)d_in[1];
  const float* nrms  = (const float*)d_in[2];
  const float* cmps  = (const float*)d_in[3];
  const int*   faces = (const int*)d_in[4];
  float* out = (float*)d_out;
  int* bestFace = (int*)d_ws;   // B*Q ints of scratch

  stage1_closest<<<(BATCH * NQ) / 16, 256, 0, stream>>>(tris, pts, bestFace);
  stage2_gather<<<(BATCH * NQ + 255) / 256, 256, 0, stream>>>(
      tris, pts, nrms, cmps, faces, bestFace, out);
}